// BiGraphContrastLayer_38517266710702
// MI455X (gfx1250) — compile-verified
//
#include <hip/hip_runtime.h>
#include <math.h>

typedef float v2f __attribute__((ext_vector_type(2)));
typedef float v8f __attribute__((ext_vector_type(8)));

#define HEADS 8
#define DIM 16
#define F 128            // H*D == IN == 128
#define NEG_SLOPE 0.2f
#define TEM 0.7f

// ---------- ordered-uint encoding for float atomic max ----------
__device__ __forceinline__ unsigned fenc(float f) {
    unsigned b = __float_as_uint(f);
    return (b & 0x80000000u) ? ~b : (b | 0x80000000u);
}
__device__ __forceinline__ float fdec(unsigned u) {
    unsigned b = (u & 0x80000000u) ? (u & 0x7FFFFFFFu) : ~u;
    return __uint_as_float(b);
}

// ---------- 1) z = feat @ fc_w via V_WMMA_F32_16X16X4_F32 ----------
// One block = one 16-row tile of feat (staged in LDS); 8 waves cover the 8
// 16-col tiles of the 128-wide output. K loop: 32 WMMAs of 16x16x4.
__global__ __launch_bounds__(256) void gemm_z_wmma(
    const float* __restrict__ feat, const float* __restrict__ fcw,
    float* __restrict__ z, int N)
{
    __shared__ float sA[16 * F];                    // 8 KB
    const int row0 = blockIdx.x * 16;

    for (int i = threadIdx.x; i < 16 * F; i += 256) {
        int r = i >> 7, c = i & 127;
        int gr = row0 + r;
        sA[i] = (gr < N) ? feat[(long long)gr * F + c] : 0.0f;
    }
    __syncthreads();

    const int wave = threadIdx.x >> 5;              // 0..7 -> col tile
    const int lane = threadIdx.x & 31;
    const int col0 = wave * 16;
    const int m    = lane & 15;                     // row (A) / col (B,C)
    const int koff = (lane >> 4) * 2;               // K sub-offset 0 or 2

    v8f c = {};
    for (int kb = 0; kb < F; kb += 4) {
        const int k = kb + koff;
        v2f a, b;
        a.x = sA[m * F + k];
        a.y = sA[m * F + k + 1];
        b.x = fcw[(long long)k * F + col0 + m];
        b.y = fcw[(long long)(k + 1) * F + col0 + m];
        c = __builtin_amdgcn_wmma_f32_16x16x4_f32(
                false, a, false, b, (short)0, c, false, false);
    }

    // C/D layout: VGPR i -> row (i + (lane>>4)*8), col (lane&15)
    for (int i = 0; i < 8; ++i) {
        int gr = row0 + i + (lane >> 4) * 8;
        if (gr < N) z[(long long)gr * F + col0 + m] = c[i];
    }
}

// ---------- 2) el/er = sum_d z[n,h,d] * attn_{l,r}[h,d] ----------
__global__ void el_er_kernel(const float* __restrict__ z,
                             const float* __restrict__ attn_l,
                             const float* __restrict__ attn_r,
                             float* __restrict__ el, float* __restrict__ er,
                             int N)
{
    int t = blockIdx.x * blockDim.x + threadIdx.x;
    if (t >= N * HEADS) return;
    int n = t >> 3, h = t & 7;
    const float* zp = z + (long long)n * F + h * DIM;
    const float* al = attn_l + h * DIM;
    const float* ar = attn_r + h * DIM;
    float sl = 0.f, sr = 0.f;
    #pragma unroll
    for (int d = 0; d < DIM; ++d) { float v = zp[d]; sl += v * al[d]; sr += v * ar[d]; }
    el[t] = sl; er[t] = sr;
}

// ---------- 3) per-graph init: rst=0, denom=0, menc=0, red=0 ----------
__global__ void init_kernel(float* __restrict__ rst, float* __restrict__ denom,
                            unsigned* __restrict__ menc, float* __restrict__ red,
                            int N)
{
    int t = blockIdx.x * blockDim.x + threadIdx.x;
    if (t < N * F) rst[t] = 0.0f;
    if (t < N * HEADS) { denom[t] = 0.0f; menc[t] = 0u; }
    if (t == 0) red[0] = 0.0f;
}

__device__ __forceinline__ void edge_lookup(const int* __restrict__ esrc,
                                            const int* __restrict__ edst,
                                            const int* __restrict__ sel,
                                            int n_real, int e, int& s, int& d)
{
    if (e < n_real) {
        int j = sel ? sel[e] : e;
        s = esrc[j]; d = edst[j];
    } else {
        s = d = e - n_real;                         // self loop
    }
}

// ---------- 4) pass 1: segment max of leaky-relu(el[src]+er[dst]) ----------
__global__ void edge_max_kernel(const float* __restrict__ el,
                                const float* __restrict__ er,
                                const int* __restrict__ esrc,
                                const int* __restrict__ edst,
                                const int* __restrict__ sel,
                                int n_real, int N,
                                unsigned* __restrict__ menc)
{
    int t = blockIdx.x * blockDim.x + threadIdx.x;
    int total = (n_real + N) * HEADS;
    if (t >= total) return;
    int e = t >> 3, h = t & 7;
    int s, d; edge_lookup(esrc, edst, sel, n_real, e, s, d);
    float x = el[s * HEADS + h] + er[d * HEADS + h];
    x = (x >= 0.f) ? x : NEG_SLOPE * x;
    atomicMax(&menc[d * HEADS + h], fenc(x));
}

// ---------- 5) pass 2: w = exp(e - m); denom += w; rst += w * z[src] ----------
__global__ void edge_acc_kernel(const float* __restrict__ el,
                                const float* __restrict__ er,
                                const float* __restrict__ z,
                                const int* __restrict__ esrc,
                                const int* __restrict__ edst,
                                const int* __restrict__ sel,
                                int n_real, int N,
                                const unsigned* __restrict__ menc,
                                float* __restrict__ denom,
                                float* __restrict__ rst)
{
    int t = blockIdx.x * blockDim.x + threadIdx.x;
    int total = (n_real + N) * HEADS;
    if (t >= total) return;
    int e = t >> 3, h = t & 7;
    int s, d; edge_lookup(esrc, edst, sel, n_real, e, s, d);
    float x = el[s * HEADS + h] + er[d * HEADS + h];
    x = (x >= 0.f) ? x : NEG_SLOPE * x;
    float mx = fdec(menc[d * HEADS + h]);
    float w = __expf(x - mx);
    atomicAdd(&denom[d * HEADS + h], w);
    const float* zp = z + (long long)s * F + h * DIM;
    float* rp = rst + (long long)d * F + h * DIM;
    #pragma unroll
    for (int k = 0; k < DIM; ++k) atomicAdd(&rp[k], w * zp[k]);
}

// ---------- 6) finalize: rst/denom + bias, PReLU, head mean ----------
__global__ void finalize_kernel(const float* __restrict__ rst,
                                const float* __restrict__ denom,
                                const float* __restrict__ bias,
                                const float* __restrict__ prelu_a,
                                float* __restrict__ hout,
                                int n0, int nCnt)
{
    int t = blockIdx.x * blockDim.x + threadIdx.x;
    if (t >= nCnt * DIM) return;
    int i = t >> 4, dcol = t & 15;
    int n = n0 + i;
    float a = prelu_a[0];
    float acc = 0.f;
    #pragma unroll
    for (int h = 0; h < HEADS; ++h) {
        float v = rst[(long long)n * F + h * DIM + dcol] / denom[n * HEADS + h]
                + bias[h * DIM + dcol];
        v = (v >= 0.f) ? v : a * v;
        acc += v;
    }
    hout[(long long)i * DIM + dcol] = acc * (1.0f / HEADS);
}

// ---------- 7) cosine + sum of exp(cos/TEM) ----------
__global__ __launch_bounds__(256) void cos_loss_kernel(
    const float* __restrict__ hp, const float* __restrict__ hn,
    float* __restrict__ red, int nCnt)
{
    int i = blockIdx.x * blockDim.x + threadIdx.x;
    float v = 0.f;
    if (i < nCnt) {
        float num = 0.f, na = 0.f, nb = 0.f;
        #pragma unroll
        for (int d = 0; d < DIM; ++d) {
            float a = hp[(long long)i * DIM + d];
            float b = hn[(long long)i * DIM + d];
            num += a * b; na += a * a; nb += b * b;
        }
        na = fmaxf(sqrtf(na), 1e-8f);
        nb = fmaxf(sqrtf(nb), 1e-8f);
        v = __expf((num / (na * nb)) * (1.0f / TEM));
    }
    __shared__ float sbuf[256];
    sbuf[threadIdx.x] = v;
    __syncthreads();
    for (int s = 128; s > 0; s >>= 1) {
        if (threadIdx.x < s) sbuf[threadIdx.x] += sbuf[threadIdx.x + s];
        __syncthreads();
    }
    if (threadIdx.x == 0) atomicAdd(red, sbuf[0]);
}

__global__ void write_loss_kernel(const float* __restrict__ red, float* __restrict__ out)
{
    out[0] = logf(red[0]);
}

// ---------------------------------------------------------------------------
extern "C" void kernel_launch(void* const* d_in, const int* in_sizes, int n_in,
                              void* d_out, int out_size, void* d_ws, size_t ws_size,
                              hipStream_t stream)
{
    const float* feat    = (const float*)d_in[0];
    const float* fcw     = (const float*)d_in[1];
    const float* attn_l  = (const float*)d_in[2];
    const float* attn_r  = (const float*)d_in[3];
    const float* bias    = (const float*)d_in[4];
    const float* prelu_a = (const float*)d_in[5];
    const int*   esrc    = (const int*)d_in[6];
    const int*   edst    = (const int*)d_in[7];
    const int*   negidx  = (const int*)d_in[8];

    const int N     = in_sizes[0] / F;          // 40000
    const int E     = in_sizes[6];              // 320000
    const int E_NEG = in_sizes[8];              // 316800
    const int N_DST = (out_size - 1) / DIM;     // 20000
    const int N_SRC = N - N_DST;

    float* out   = (float*)d_out;
    float* h_pos = out + 1;                     // (N_DST, 16)

    // workspace layout (floats)
    float*    ws    = (float*)d_ws;
    float*    z     = ws;                       // N*F
    float*    el    = z   + (size_t)N * F;      // N*H
    float*    er    = el  + (size_t)N * HEADS;  // N*H
    float*    denom = er  + (size_t)N * HEADS;  // N*H
    unsigned* menc  = (unsigned*)(denom + (size_t)N * HEADS); // N*H
    float*    rst   = (float*)(menc + (size_t)N * HEADS);     // N*F
    float*    h_neg = rst + (size_t)N * F;      // N_DST*16
    float*    red   = h_neg + (size_t)N_DST * DIM;            // 1

    const int TB = 256;
    auto cdiv = [](long long a, int b) { return (unsigned)((a + b - 1) / b); };

    // 1) projection GEMM (WMMA)
    gemm_z_wmma<<<cdiv(N, 16), TB, 0, stream>>>(feat, fcw, z, N);

    // 2) attention logits per node/head
    el_er_kernel<<<cdiv((long long)N * HEADS, TB), TB, 0, stream>>>(
        z, attn_l, attn_r, el, er, N);

    // ---- graph A: all edges -> h_pos (written straight into d_out) ----
    init_kernel<<<cdiv((long long)N * F, TB), TB, 0, stream>>>(rst, denom, menc, red, N);
    edge_max_kernel<<<cdiv((long long)(E + N) * HEADS, TB), TB, 0, stream>>>(
        el, er, esrc, edst, nullptr, E, N, menc);
    edge_acc_kernel<<<cdiv((long long)(E + N) * HEADS, TB), TB, 0, stream>>>(
        el, er, z, esrc, edst, nullptr, E, N, menc, denom, rst);
    finalize_kernel<<<cdiv((long long)N_DST * DIM, TB), TB, 0, stream>>>(
        rst, denom, bias, prelu_a, h_pos, N_SRC, N_DST);

    // ---- graph B: neg-sampled edges -> h_neg ----
    init_kernel<<<cdiv((long long)N * F, TB), TB, 0, stream>>>(rst, denom, menc, red, N);
    edge_max_kernel<<<cdiv((long long)(E_NEG + N) * HEADS, TB), TB, 0, stream>>>(
        el, er, esrc, edst, negidx, E_NEG, N, menc);
    edge_acc_kernel<<<cdiv((long long)(E_NEG + N) * HEADS, TB), TB, 0, stream>>>(
        el, er, z, esrc, edst, negidx, E_NEG, N, menc, denom, rst);
    finalize_kernel<<<cdiv((long long)N_DST * DIM, TB), TB, 0, stream>>>(
        rst, denom, bias, prelu_a, h_neg, N_SRC, N_DST);

    // ---- contrastive loss ----
    cos_loss_kernel<<<cdiv(N_DST, TB), TB, 0, stream>>>(h_pos, h_neg, red, N_DST);
    write_loss_kernel<<<1, 1, 0, stream>>>(red, out);
}